// GTR_18245021073878
// MI455X (gfx1250) — compile-verified
//
#include <hip/hip_runtime.h>
#include <math.h>

typedef __attribute__((ext_vector_type(2))) float v2f;
typedef __attribute__((ext_vector_type(8))) float v8f;

#define NSTATES 64
#define BATCH   32768
#define NSWEEPS 14

// ---------------------------------------------------------------------------
// Kernel 1: build symmetrized rate matrix S, Jacobi eigendecomposition,
// emit evals[64], A[64][64] (=diag(1/sqrt(pi))*V), Bt[64][64] (=diag(sqrt(pi))*V)
// into workspace. One block of 64 threads.
// ---------------------------------------------------------------------------
__device__ __forceinline__ float softplus_f(float x) {
    return (x > 20.0f) ? x : log1pf(__expf(x));
}

__global__ void gtr_setup_eigh(const float* __restrict__ logits,
                               const float* __restrict__ E,
                               float* __restrict__ ws) {
    __shared__ float S[NSTATES][NSTATES + 1];
    __shared__ float V[NSTATES][NSTATES + 1];
    __shared__ float pi[NSTATES], spi[NSTATES], ispi[NSTATES];
    __shared__ float cc[32], ss[32];
    __shared__ int   pp[32], qq[32];

    const int tid = threadIdx.x;  // 0..63

    // --- softmax(logits) -> pi (tiny; serialize on thread 0) ---
    if (tid < NSTATES) pi[tid] = logits[tid];
    __syncthreads();
    if (tid == 0) {
        float m = -3.4e38f;
        for (int j = 0; j < NSTATES; ++j) m = fmaxf(m, pi[j]);
        float sum = 0.0f;
        for (int j = 0; j < NSTATES; ++j) { float e = __expf(pi[j] - m); pi[j] = e; sum += e; }
        float inv = 1.0f / sum;
        for (int j = 0; j < NSTATES; ++j) pi[j] *= inv;
    }
    __syncthreads();
    spi[tid]  = sqrtf(pi[tid]);
    ispi[tid] = 1.0f / spi[tid];
    __syncthreads();

    // --- build symmetrized S row i = tid; init V = I ---
    {
        const int i = tid;
        float rowsum = 0.0f;
        for (int j = 0; j < NSTATES; ++j) {
            V[i][j] = (i == j) ? 1.0f : 0.0f;
            if (j == i) continue;
            float e = (i < j) ? E[i * NSTATES + j] : E[j * NSTATES + i];
            float q = softplus_f(e) * pi[j];    // Q_ij (off-diag)
            rowsum += q;
            S[i][j] = spi[i] * q * ispi[j];     // = R_ij * sqrt(pi_i*pi_j), symmetric
        }
        S[i][i] = -rowsum;                      // scale factors cancel on diagonal
    }
    __syncthreads();

    // --- parallel cyclic Jacobi: 63 tournament rounds x 32 disjoint pairs ---
    for (int sweep = 0; sweep < NSWEEPS; ++sweep) {
        for (int r = 0; r < 63; ++r) {
            if (tid < 32) {
                int m = tid, p, q;
                if (m == 0) { p = 63; q = r; }
                else        { p = (r + m) % 63; q = (r - m + 63) % 63; }
                if (p > q) { int t = p; p = q; q = t; }
                float app = S[p][p], aqq = S[q][q], apq = S[p][q];
                float c = 1.0f, s = 0.0f;
                if (fabsf(apq) > 1e-12f) {
                    float tau = (aqq - app) / (2.0f * apq);
                    float t   = ((tau >= 0.0f) ? 1.0f : -1.0f) /
                                (fabsf(tau) + sqrtf(1.0f + tau * tau));
                    c = 1.0f / sqrtf(1.0f + t * t);
                    s = t * c;
                }
                pp[m] = p; qq[m] = q; cc[m] = c; ss[m] = s;
            }
            __syncthreads();
            // row update: thread owns column j = tid (rows of each pair disjoint)
            {
                const int j = tid;
                #pragma unroll 4
                for (int m = 0; m < 32; ++m) {
                    int p = pp[m], q = qq[m];
                    float c = cc[m], s = ss[m];
                    float Sp = S[p][j], Sq = S[q][j];
                    S[p][j] = c * Sp - s * Sq;
                    S[q][j] = s * Sp + c * Sq;
                }
            }
            __syncthreads();
            // column update (+ eigenvector accumulation): thread owns row i = tid
            {
                const int i = tid;
                #pragma unroll 4
                for (int m = 0; m < 32; ++m) {
                    int p = pp[m], q = qq[m];
                    float c = cc[m], s = ss[m];
                    float Sip = S[i][p], Siq = S[i][q];
                    S[i][p] = c * Sip - s * Siq;
                    S[i][q] = s * Sip + c * Siq;
                    float Vip = V[i][p], Viq = V[i][q];
                    V[i][p] = c * Vip - s * Viq;
                    V[i][q] = s * Vip + c * Viq;
                }
            }
            __syncthreads();
        }
    }

    // --- emit: ws = [evals(64) | A(4096) | Bt(4096)] ---
    float* evals = ws;
    float* A     = ws + NSTATES;
    float* Bt    = ws + NSTATES + NSTATES * NSTATES;
    evals[tid] = S[tid][tid];
    const int i = tid;
    for (int k = 0; k < NSTATES; ++k) {
        float v = V[i][k];
        A[i * NSTATES + k]  = ispi[i] * v;   // rows of left factor
        Bt[i * NSTATES + k] = spi[i]  * v;   // rows of right factor (used transposed)
    }
}

// ---------------------------------------------------------------------------
// Kernel 2: batched P_b = A * diag(exp(evals*t_b)) * Bt^T, then log-clamp.
// 128 threads = 4 waves; wave w owns output rows [16w, 16w+16); grid-stride b.
// WMMA: V_WMMA_F32_16X16X4_F32, 64 per wave per batch element.
// ---------------------------------------------------------------------------
__global__ void gtr_batched_expm(const float* __restrict__ t,
                                 const float* __restrict__ ws,
                                 float* __restrict__ out) {
    __shared__ float Alds[NSTATES * NSTATES];   // 16 KB
    __shared__ float Blds[NSTATES * NSTATES];   // 16 KB
    __shared__ float ev[NSTATES];
    __shared__ float w[NSTATES];

    const float* Ag  = ws + NSTATES;
    const float* Btg = ws + NSTATES + NSTATES * NSTATES;

    const int tid = threadIdx.x;
    for (int idx = tid; idx < NSTATES * NSTATES; idx += 128) {
        Alds[idx] = Ag[idx];
        Blds[idx] = Btg[idx];
    }
    if (tid < NSTATES) ev[tid] = ws[tid];
    __syncthreads();

    const int waveId = tid >> 5;        // 0..3 -> row strip
    const int lane   = tid & 31;
    const int lane15 = lane & 15;
    const int half   = lane >> 4;       // 0: lanes 0-15, 1: lanes 16-31
    const int rowA   = waveId * 16 + lane15;

    for (int b = blockIdx.x; b < BATCH; b += gridDim.x) {
        __syncthreads();                          // retire prior reads of w[]
        const float tb = t[b];
        if (tid < NSTATES) w[tid] = __expf(ev[tid] * tb);
        __syncthreads();

        v8f acc0 = {}, acc1 = {}, acc2 = {}, acc3 = {};

        #pragma unroll
        for (int kt = 0; kt < 16; ++kt) {
            const int k0 = kt * 4 + half * 2;     // A layout: v0 holds K+0|K+2, v1 holds K+1|K+3
            const float2 wv = *(const float2*)&w[k0];
            const float2 av = *(const float2*)&Alds[rowA * NSTATES + k0];
            v2f afrag; afrag.x = av.x * wv.x; afrag.y = av.y * wv.y;

            const float2 b0 = *(const float2*)&Blds[(0 * 16 + lane15) * NSTATES + k0];
            const float2 b1 = *(const float2*)&Blds[(1 * 16 + lane15) * NSTATES + k0];
            const float2 b2 = *(const float2*)&Blds[(2 * 16 + lane15) * NSTATES + k0];
            const float2 b3 = *(const float2*)&Blds[(3 * 16 + lane15) * NSTATES + k0];
            v2f f0; f0.x = b0.x; f0.y = b0.y;
            v2f f1; f1.x = b1.x; f1.y = b1.y;
            v2f f2; f2.x = b2.x; f2.y = b2.y;
            v2f f3; f3.x = b3.x; f3.y = b3.y;

            acc0 = __builtin_amdgcn_wmma_f32_16x16x4_f32(false, afrag, false, f0, (short)0, acc0, false, false);
            acc1 = __builtin_amdgcn_wmma_f32_16x16x4_f32(false, afrag, false, f1, (short)0, acc1, false, false);
            acc2 = __builtin_amdgcn_wmma_f32_16x16x4_f32(false, afrag, false, f2, (short)0, acc2, false, false);
            acc3 = __builtin_amdgcn_wmma_f32_16x16x4_f32(false, afrag, false, f3, (short)0, acc3, false, false);
        }

        // epilogue: C/D layout VGPR r -> row (r + 8*half), col = lane15
        float* ob = out + (size_t)b * (NSTATES * NSTATES);
        #pragma unroll
        for (int r = 0; r < 8; ++r) {
            const int rowo = waveId * 16 + r + half * 8;
            float* orow = ob + rowo * NSTATES + lane15;
            orow[0]  = __logf(fmaxf(acc0[r], 1e-30f));
            orow[16] = __logf(fmaxf(acc1[r], 1e-30f));
            orow[32] = __logf(fmaxf(acc2[r], 1e-30f));
            orow[48] = __logf(fmaxf(acc3[r], 1e-30f));
        }
    }
}

// ---------------------------------------------------------------------------
extern "C" void kernel_launch(void* const* d_in, const int* in_sizes, int n_in,
                              void* d_out, int out_size, void* d_ws, size_t ws_size,
                              hipStream_t stream) {
    const float* t      = (const float*)d_in[0];
    const float* logits = (const float*)d_in[1];
    const float* E      = (const float*)d_in[2];
    float*       out    = (float*)d_out;
    float*       ws     = (float*)d_ws;

    gtr_setup_eigh<<<1, 64, 0, stream>>>(logits, E, ws);
    gtr_batched_expm<<<2048, 128, 0, stream>>>(t, ws, out);
}